// LocalToGlobalRegistration_56831007261101
// MI455X (gfx1250) — compile-verified
//
#include <hip/hip_runtime.h>
#include <math.h>

// ---------------- problem constants ----------------
#define BB   32
#define KK   4096
#define LL   128
#define NSLOT 129          // 128 groups + fallback slot
static __device__ __constant__ float d_ACC2 = 0.01f;   // 0.1^2

typedef __attribute__((ext_vector_type(2))) float v2f;
typedef __attribute__((ext_vector_type(8))) float v8f;

// quad-of-4 lane sum via DPP quad_perm (no LDS, no waits)
__device__ inline float quad_sum4(float x) {
    int a = __builtin_amdgcn_update_dpp(0, __float_as_int(x), 0xB1, 0xF, 0xF, true); // [1,0,3,2]
    x += __int_as_float(a);
    int b = __builtin_amdgcn_update_dpp(0, __float_as_int(x), 0x4E, 0xF, 0xF, true); // [2,3,0,1]
    x += __int_as_float(b);
    return x;
}

// ---------------- 3x3 Kabsch helpers ----------------
__device__ inline void jrot(float A[3][3], float V[3][3], int p, int q) {
    float apq = A[p][q];
    if (fabsf(apq) < 1e-20f) return;
    float app = A[p][p], aqq = A[q][q];
    float tau = (aqq - app) / (2.0f * apq);
    float tt  = copysignf(1.0f, tau) / (fabsf(tau) + sqrtf(1.0f + tau * tau));
    float c   = rsqrtf(1.0f + tt * tt);
    float s   = tt * c;
    for (int k = 0; k < 3; ++k) {
        float akp = A[k][p], akq = A[k][q];
        A[k][p] = c * akp - s * akq;
        A[k][q] = s * akp + c * akq;
    }
    for (int k = 0; k < 3; ++k) {
        float apk = A[p][k], aqk = A[q][k];
        A[p][k] = c * apk - s * aqk;
        A[q][k] = s * apk + c * aqk;
    }
    for (int k = 0; k < 3; ++k) {
        float vkp = V[k][p], vkq = V[k][q];
        V[k][p] = c * vkp - s * vkq;
        V[k][q] = s * vkp + c * vkq;
    }
}

// stats s[16] = {S0, Ssx,Ssy,Ssz, Stx,Sty,Stz, M00..M22}; outputs R (row-major), t
__device__ inline void kabsch_from_stats(const float* s, float* R, float* t) {
    float inv = 1.0f / (s[0] + 1e-8f);
    float cs[3] = { s[1] * inv, s[2] * inv, s[3] * inv };
    float ct[3] = { s[4] * inv, s[5] * inv, s[6] * inv };
    float H[3][3];
    #pragma unroll
    for (int i = 0; i < 3; ++i)
        #pragma unroll
        for (int j = 0; j < 3; ++j)
            H[i][j] = s[7 + i * 3 + j] * inv - cs[i] * ct[j];
    H[0][0] += 1e-6f; H[1][1] += 1e-6f; H[2][2] += 1e-6f;

    float A[3][3];
    #pragma unroll
    for (int i = 0; i < 3; ++i)
        #pragma unroll
        for (int j = 0; j < 3; ++j)
            A[i][j] = H[0][i] * H[0][j] + H[1][i] * H[1][j] + H[2][i] * H[2][j];

    float V[3][3] = { {1,0,0},{0,1,0},{0,0,1} };
    #pragma unroll
    for (int sweep = 0; sweep < 8; ++sweep) {
        jrot(A, V, 0, 1); jrot(A, V, 0, 2); jrot(A, V, 1, 2);
    }
    float lam[3] = { A[0][0], A[1][1], A[2][2] };
    #pragma unroll
    for (int a = 0; a < 2; ++a)
        #pragma unroll
        for (int bidx = a + 1; bidx < 3; ++bidx)
            if (lam[a] < lam[bidx]) {
                float tl = lam[a]; lam[a] = lam[bidx]; lam[bidx] = tl;
                for (int k = 0; k < 3; ++k) { float tv = V[k][a]; V[k][a] = V[k][bidx]; V[k][bidx] = tv; }
            }
    float detV = V[0][0] * (V[1][1] * V[2][2] - V[1][2] * V[2][1])
               - V[0][1] * (V[1][0] * V[2][2] - V[1][2] * V[2][0])
               + V[0][2] * (V[1][0] * V[2][1] - V[1][1] * V[2][0]);
    if (detV < 0.0f) { V[0][2] = -V[0][2]; V[1][2] = -V[1][2]; V[2][2] = -V[2][2]; }

    float u0[3], u1[3], u2[3];
    #pragma unroll
    for (int i = 0; i < 3; ++i)
        u0[i] = H[i][0] * V[0][0] + H[i][1] * V[1][0] + H[i][2] * V[2][0];
    float n0 = sqrtf(u0[0]*u0[0] + u0[1]*u0[1] + u0[2]*u0[2]);
    if (n0 > 1e-20f) { float r = 1.0f / n0; u0[0]*=r; u0[1]*=r; u0[2]*=r; }
    else { u0[0] = 1.0f; u0[1] = 0.0f; u0[2] = 0.0f; }
    #pragma unroll
    for (int i = 0; i < 3; ++i)
        u1[i] = H[i][0] * V[0][1] + H[i][1] * V[1][1] + H[i][2] * V[2][1];
    float d01 = u0[0]*u1[0] + u0[1]*u1[1] + u0[2]*u1[2];
    u1[0] -= d01 * u0[0]; u1[1] -= d01 * u0[1]; u1[2] -= d01 * u0[2];
    float n1 = sqrtf(u1[0]*u1[0] + u1[1]*u1[1] + u1[2]*u1[2]);
    if (n1 > 1e-20f) { float r = 1.0f / n1; u1[0]*=r; u1[1]*=r; u1[2]*=r; }
    else {
        float e[3] = { (fabsf(u0[0]) < 0.9f) ? 1.0f : 0.0f,
                       (fabsf(u0[0]) < 0.9f) ? 0.0f : 1.0f, 0.0f };
        u1[0] = u0[1]*e[2] - u0[2]*e[1];
        u1[1] = u0[2]*e[0] - u0[0]*e[2];
        u1[2] = u0[0]*e[1] - u0[1]*e[0];
        float nn = rsqrtf(u1[0]*u1[0] + u1[1]*u1[1] + u1[2]*u1[2] + 1e-30f);
        u1[0]*=nn; u1[1]*=nn; u1[2]*=nn;
    }
    u2[0] = u0[1]*u1[2] - u0[2]*u1[1];
    u2[1] = u0[2]*u1[0] - u0[0]*u1[2];
    u2[2] = u0[0]*u1[1] - u0[1]*u1[0];

    #pragma unroll
    for (int i = 0; i < 3; ++i)
        #pragma unroll
        for (int j = 0; j < 3; ++j)
            R[i * 3 + j] = V[i][0] * u0[j] + V[i][1] * u1[j] + V[i][2] * u2[j];
    #pragma unroll
    for (int i = 0; i < 3; ++i)
        t[i] = ct[i] - (R[i*3+0] * cs[0] + R[i*3+1] * cs[1] + R[i*3+2] * cs[2]);
}

// ---------------- kernel 1: segmented stats ----------------
__global__ __launch_bounds__(256)
void k_accum(const float* __restrict__ src, const float* __restrict__ tgt,
             const float* __restrict__ w, const int* __restrict__ labels,
             float* __restrict__ stats) {
    int b = blockIdx.x, tid = threadIdx.x;
    __shared__ float sm[NSLOT * 16];
    for (int i = tid; i < NSLOT * 16; i += 256) sm[i] = 0.0f;
    __syncthreads();
    for (int k = tid; k < KK; k += 256) {
        float wk = w[(size_t)b * KK + k];
        if (wk > 0.5f) {
            int g = labels[(size_t)b * KK + k];
            const float* sp = src + ((size_t)b * KK + k) * 3;
            const float* tp = tgt + ((size_t)b * KK + k) * 3;
            float sx = sp[0], sy = sp[1], sz = sp[2];
            float tx = tp[0], ty = tp[1], tz = tp[2];
            float v[16] = { 1.0f, sx, sy, sz, tx, ty, tz,
                            sx*tx, sx*ty, sx*tz, sy*tx, sy*ty, sy*tz,
                            sz*tx, sz*ty, sz*tz };
            float* g0 = sm + g * 16;
            float* g1 = sm + 128 * 16;
            #pragma unroll
            for (int i = 0; i < 16; ++i) { atomicAdd(g0 + i, v[i]); atomicAdd(g1 + i, v[i]); }
        }
    }
    __syncthreads();
    for (int i = tid; i < NSLOT * 16; i += 256)
        stats[(size_t)b * NSLOT * 16 + i] = sm[i];
}

// ---------------- kernel 2: per-group Kabsch ----------------
__global__ __launch_bounds__(128)
void k_kabsch(const float* __restrict__ stats, float* __restrict__ xforms,
              int* __restrict__ gflag) {
    int idx = blockIdx.x * blockDim.x + threadIdx.x;
    if (idx >= BB * NSLOT) return;
    const float* s = stats + (size_t)idx * 16;
    float* xf = xforms + (size_t)idx * 12;
    float S0 = s[0];
    gflag[idx] = (S0 >= 3.0f) ? 1 : 0;
    if (S0 < 0.5f) {
        xf[0]=1; xf[1]=0; xf[2]=0; xf[3]=0; xf[4]=1; xf[5]=0; xf[6]=0; xf[7]=0; xf[8]=1;
        xf[9]=0; xf[10]=0; xf[11]=0;
        return;
    }
    float R[9], t[3];
    kabsch_from_stats(s, R, t);
    #pragma unroll
    for (int i = 0; i < 9; ++i) xf[i] = R[i];
    xf[9] = t[0]; xf[10] = t[1]; xf[11] = t[2];
}

// ---------------- kernel 3: WMMA hypothesis verification ----------------
// D = A(16x4)*B(4x16) + C : A = [sx sy sz 1], each group owns 4 padded columns
// [R_row_c | t_c] (c=0..2, col 3 dead), C = -tgt[m][comp] so D IS the residual
// component. comp = lane&3 is invariant across column tiles, so C loads once
// per row tile and is reused by all 32 WMMAs. Column loop FULLY unrolled so B
// fragments + counters stay in VGPRs (no scratch, no loads in the hot loop).
__global__ __launch_bounds__(256)
void k_count(const float* __restrict__ src, const float* __restrict__ tgt,
             const float* __restrict__ w, const float* __restrict__ xforms,
             const int* __restrict__ gflag, int* __restrict__ counts) {
    int b = blockIdx.x, tid = threadIdx.x;
    __shared__ float Wlds[512 * 4];          // col-major: [col][row0..3]
    __shared__ unsigned vmask[KK / 32];
    __shared__ int cnt[LL];

    for (int col = tid; col < 512; col += 256) {
        int g = col >> 2, c = col & 3;
        float r0 = 0.f, r1 = 0.f, r2 = 0.f, r3 = 0.f;
        if (c < 3) {
            const float* xf = xforms + ((size_t)b * NSLOT + g) * 12;
            r0 = xf[c * 3 + 0]; r1 = xf[c * 3 + 1]; r2 = xf[c * 3 + 2]; r3 = xf[9 + c];
        }
        Wlds[col * 4 + 0] = r0; Wlds[col * 4 + 1] = r1;
        Wlds[col * 4 + 2] = r2; Wlds[col * 4 + 3] = r3;
    }
    for (int i = tid; i < KK / 32; i += 256) {
        unsigned m = 0;
        const float* wp = w + (size_t)b * KK + i * 32;
        #pragma unroll
        for (int j = 0; j < 32; ++j) m |= (wp[j] > 0.5f ? 1u : 0u) << j;
        vmask[i] = m;
    }
    for (int g = tid; g < LL; g += 256) cnt[g] = 0;
    __syncthreads();

    int wave = tid >> 5, lane = tid & 31;
    int hi = lane >> 4;                 // 0: A K=0,1 / D rows r ; 1: A K=2,3 / D rows r+8
    int n = lane & 15;
    const float* srcB = src + (size_t)b * KK * 3;
    const float* tgtB = tgt + (size_t)b * KK * 3;
    float acc2 = d_ACC2;

    // all 32 B fragments in registers (constant indices -> VGPRs, not scratch)
    v2f bfr[32];
    #pragma unroll
    for (int ct = 0; ct < 32; ++ct) {
        const float* wc = &Wlds[(ct * 16 + n) * 4];
        int o = hi ? 2 : 0;
        bfr[ct].x = wc[o + 0];
        bfr[ct].y = wc[o + 1];
    }

    int comp = lane & 3;
    int comp3 = (comp < 3) ? comp : 0;      // clamped (branchless load)
    float csel = (comp < 3) ? -1.0f : 0.0f; // zero out dead column

    int cnt32[32];
    #pragma unroll
    for (int ct = 0; ct < 32; ++ct) cnt32[ct] = 0;

    for (int rt = wave; rt < KK / 16; rt += 8) {   // uniform across the 8 waves
        int rowBase = rt * 16;
        const float* sp = srcB + (size_t)(rowBase + n) * 3;
        v2f a;
        int o0 = hi ? 2 : 0;
        float s1 = sp[1];
        a.x = sp[o0];
        a.y = hi ? 1.0f : s1;

        int rowOff = rowBase + (hi ? 8 : 0);
        union { v8f v; float f[8]; } cu;
        #pragma unroll
        for (int r = 0; r < 8; ++r)
            cu.f[r] = csel * tgtB[(size_t)(rowOff + r) * 3 + comp3];

        unsigned mbits = (vmask[rowOff >> 5] >> (rowOff & 31)) & 0xFFu;

        #pragma unroll
        for (int ct = 0; ct < 32; ++ct) {
            v8f d = __builtin_amdgcn_wmma_f32_16x16x4_f32(
                        false, a, false, bfr[ct], (short)0, cu.v, false, false);
            union { v8f v; float f[8]; } du; du.v = d;
            #pragma unroll
            for (int r = 0; r < 8; ++r) {
                float e = du.f[r];
                float e2 = quad_sum4(e * e);       // sum 3 comps (+0) in quad
                bool val = (mbits >> r) & 1u;
                cnt32[ct] += (val && e2 < acc2) ? 1 : 0;
            }
        }
    }

    if (comp == 0) {
        #pragma unroll
        for (int ct = 0; ct < 32; ++ct)
            if (cnt32[ct]) atomicAdd(&cnt[ct * 4 + (n >> 2)], cnt32[ct]);
    }
    __syncthreads();
    for (int g = tid; g < LL; g += 256) {
        int flag = gflag[(size_t)b * NSLOT + g];
        counts[(size_t)b * LL + g] = flag ? cnt[g] : -1;
    }
}

// ---------------- kernel 4: select + refine + output ----------------
__global__ __launch_bounds__(256)
void k_refine(const float* __restrict__ src, const float* __restrict__ tgt,
              const float* __restrict__ w, const float* __restrict__ stats,
              const float* __restrict__ xforms, const int* __restrict__ counts,
              float* __restrict__ out) {
    int b = blockIdx.x, tid = threadIdx.x;
    __shared__ float Rt[12];
    __shared__ float acc[16];

    if (tid == 0) {
        int best = 0, bv = counts[(size_t)b * LL];
        for (int g = 1; g < LL; ++g) {
            int v = counts[(size_t)b * LL + g];
            if (v > bv) { bv = v; best = g; }      // first-max (strict greater)
        }
        int slot = (bv <= 0) ? 128 : best;         // fallback = all-valid Kabsch
        const float* xf = xforms + ((size_t)b * NSLOT + slot) * 12;
        for (int i = 0; i < 12; ++i) Rt[i] = xf[i];
    }
    __syncthreads();

    float acc2 = d_ACC2;
    for (int it = 0; it < 5; ++it) {
        if (tid < 16) acc[tid] = 0.0f;
        __syncthreads();
        float R00 = Rt[0], R01 = Rt[1], R02 = Rt[2];
        float R10 = Rt[3], R11 = Rt[4], R12 = Rt[5];
        float R20 = Rt[6], R21 = Rt[7], R22 = Rt[8];
        float t0 = Rt[9], t1 = Rt[10], t2 = Rt[11];
        float ls[16];
        #pragma unroll
        for (int i = 0; i < 16; ++i) ls[i] = 0.0f;
        for (int k = tid; k < KK; k += 256) {
            float wk = w[(size_t)b * KK + k];
            if (wk > 0.5f) {
                const float* sp = src + ((size_t)b * KK + k) * 3;
                const float* tp = tgt + ((size_t)b * KK + k) * 3;
                float sx = sp[0], sy = sp[1], sz = sp[2];
                float tx = tp[0], ty = tp[1], tz = tp[2];
                float dx = R00 * sx + R01 * sy + R02 * sz + t0 - tx;
                float dy = R10 * sx + R11 * sy + R12 * sz + t1 - ty;
                float dz = R20 * sx + R21 * sy + R22 * sz + t2 - tz;
                if (dx * dx + dy * dy + dz * dz < acc2) {
                    ls[0] += 1.0f;
                    ls[1] += sx; ls[2] += sy; ls[3] += sz;
                    ls[4] += tx; ls[5] += ty; ls[6] += tz;
                    ls[7] += sx * tx; ls[8] += sx * ty; ls[9] += sx * tz;
                    ls[10] += sy * tx; ls[11] += sy * ty; ls[12] += sy * tz;
                    ls[13] += sz * tx; ls[14] += sz * ty; ls[15] += sz * tz;
                }
            }
        }
        if (ls[0] > 0.0f) {
            #pragma unroll
            for (int i = 0; i < 16; ++i) atomicAdd(&acc[i], ls[i]);
        }
        __syncthreads();
        if (tid == 0 && acc[0] >= 3.0f) {
            float R[9], t[3];
            kabsch_from_stats(acc, R, t);
            for (int i = 0; i < 9; ++i) Rt[i] = R[i];
            Rt[9] = t[0]; Rt[10] = t[1]; Rt[11] = t[2];
        }
        __syncthreads();
    }

    if (tid == 0) {
        float enough = stats[((size_t)b * NSLOT + 128) * 16 + 0];
        float Rf[12];
        if (enough >= 3.0f) { for (int i = 0; i < 12; ++i) Rf[i] = Rt[i]; }
        else {
            Rf[0]=1; Rf[1]=0; Rf[2]=0; Rf[3]=0; Rf[4]=1; Rf[5]=0; Rf[6]=0; Rf[7]=0; Rf[8]=1;
            Rf[9]=0; Rf[10]=0; Rf[11]=0;
        }
        for (int i = 0; i < 9; ++i) out[(size_t)b * 9 + i] = Rf[i];
        out[(size_t)BB * 9 + b * 3 + 0] = Rf[9];
        out[(size_t)BB * 9 + b * 3 + 1] = Rf[10];
        out[(size_t)BB * 9 + b * 3 + 2] = Rf[11];
    }
}

// ---------------- launcher ----------------
extern "C" void kernel_launch(void* const* d_in, const int* in_sizes, int n_in,
                              void* d_out, int out_size, void* d_ws, size_t ws_size,
                              hipStream_t stream) {
    const float* src    = (const float*)d_in[0];   // (32,4096,3)
    const float* tgt    = (const float*)d_in[1];   // (32,4096,3)
    const float* wts    = (const float*)d_in[2];   // (32,4096)
    // d_in[3] = source_points: unused by the reference
    const int*   labels = (const int*)d_in[4];     // (32,4096)
    float* out = (float*)d_out;                    // 32*9 R then 32*3 t

    float* stats  = (float*)d_ws;                               // 32*129*16 f
    float* xforms = stats + (size_t)BB * NSLOT * 16;            // 32*129*12 f
    int*   gflag  = (int*)(xforms + (size_t)BB * NSLOT * 12);   // 32*129 i
    int*   counts = gflag + (size_t)BB * NSLOT;                 // 32*128 i

    k_accum<<<BB, 256, 0, stream>>>(src, tgt, wts, labels, stats);
    k_kabsch<<<(BB * NSLOT + 127) / 128, 128, 0, stream>>>(stats, xforms, gflag);
    k_count<<<BB, 256, 0, stream>>>(src, tgt, wts, xforms, gflag, counts);
    k_refine<<<BB, 256, 0, stream>>>(src, tgt, wts, stats, xforms, counts, out);
}